// QwenVL_PartF_20968030339732
// MI455X (gfx1250) — compile-verified
//
#include <hip/hip_runtime.h>
#include <stdint.h>

// ---------------- problem constants ----------------
#define D      2048
#define HQ     16
#define HDIM   128
#define HKV    2
#define GQ     8          // q heads per kv head
#define II     5504
#define VOCAB  32000
#define HIST   8192
#define KVLEN  8193
#define KVP    8208       // 513*16, KV padded to WMMA tile multiple
#define KLIM   8196       // KVLEN rounded up to 4 (WMMA K step)
#define NLAYER 4
#define SCALEF 0.29730177875068026f   // 128^-0.25

typedef __attribute__((ext_vector_type(2))) float v2f;
typedef __attribute__((ext_vector_type(8))) float v8f;

// Only check for the device builtin during the DEVICE pass; the host pass
// targets x86 where __has_builtin(...) is legitimately 0.
#if defined(__HIP_DEVICE_COMPILE__) && !__has_builtin(__builtin_amdgcn_wmma_f32_16x16x4_f32)
#error "wmma f32 16x16x4 builtin not available on gfx1250 device pass"
#endif

// WMMA wrapper: real builtin on device pass, parse-safe stub on host pass.
__device__ inline v8f wmma_f32_16x16x4(v2f a, v2f b, v8f c) {
#if defined(__HIP_DEVICE_COMPILE__)
  return __builtin_amdgcn_wmma_f32_16x16x4_f32(false, a, false, b, (short)0, c, false, false);
#else
  (void)a; (void)b;
  return c;
#endif
}

// ---------------- reductions (wave32) ----------------
__device__ inline float wave_reduce_sum(float v) {
#pragma unroll
  for (int o = 16; o; o >>= 1) v += __shfl_xor(v, o, 32);
  return v;
}

__device__ inline float block_reduce_sum(float v, float* sm) {
  __syncthreads();
  v = wave_reduce_sum(v);
  int wid = threadIdx.x >> 5, lane = threadIdx.x & 31;
  if (lane == 0) sm[wid] = v;
  __syncthreads();
  if (threadIdx.x == 0) {
    float s = 0.f;
    int nw = (int)(blockDim.x >> 5);
    for (int i = 0; i < nw; ++i) s += sm[i];
    sm[0] = s;
  }
  __syncthreads();
  return sm[0];
}

__device__ inline float block_reduce_max(float v, float* sm) {
  __syncthreads();
#pragma unroll
  for (int o = 16; o; o >>= 1) v = fmaxf(v, __shfl_xor(v, o, 32));
  int wid = threadIdx.x >> 5, lane = threadIdx.x & 31;
  if (lane == 0) sm[wid] = v;
  __syncthreads();
  if (threadIdx.x == 0) {
    float s = sm[0];
    int nw = (int)(blockDim.x >> 5);
    for (int i = 1; i < nw; ++i) s = fmaxf(s, sm[i]);
    sm[0] = s;
  }
  __syncthreads();
  return sm[0];
}

// ---------------- small helpers ----------------
__global__ void copy_kernel(const float* __restrict__ src, float* __restrict__ dst, int n) {
  int i = blockIdx.x * blockDim.x + threadIdx.x;
  if (i < n) dst[i] = src[i];
}

__global__ void zero_kernel(float* __restrict__ p, int n) {
  int i = blockIdx.x * blockDim.x + threadIdx.x;
  if (i < n) p[i] = 0.f;
}

// past_k [L,HKV,HD,HIST] -> new_k [L,HKV,HD,KVLEN] (cols 0..HIST-1)
__global__ void copy_k_kernel(const float* __restrict__ src, float* __restrict__ dst) {
  size_t n = (size_t)NLAYER * HKV * HDIM * HIST;
  for (size_t i = (size_t)blockIdx.x * blockDim.x + threadIdx.x; i < n;
       i += (size_t)gridDim.x * blockDim.x) {
    size_t row = i >> 13;          // (l,kv,d)
    size_t c   = i & (HIST - 1);
    dst[row * KVLEN + c] = src[i];
  }
}

// past_v [L,HKV,HIST,HD] -> new_v [L,HKV,KVLEN,HD] (rows 0..HIST-1), float4
__global__ void copy_v_kernel(const float4* __restrict__ src, float4* __restrict__ dst) {
  size_t n = (size_t)NLAYER * HKV * HIST * (HDIM / 4);
  for (size_t i = (size_t)blockIdx.x * blockDim.x + threadIdx.x; i < n;
       i += (size_t)gridDim.x * blockDim.x) {
    size_t d4  = i & 31;
    size_t row = i >> 5;           // (l,kv,c)
    size_t lkv = row >> 13;
    size_t c   = row & (HIST - 1);
    dst[(lkv * KVLEN + c) * 32 + d4] = src[i];
  }
}

// ---------------- RMSNorm ----------------
__global__ void rmsnorm_kernel(const float* __restrict__ x, const float* __restrict__ w,
                               float* __restrict__ out) {
  __shared__ float sm[8];
  float s = 0.f;
  for (int i = threadIdx.x; i < D; i += blockDim.x) { float v = x[i]; s += v * v; }
  s = block_reduce_sum(s, sm);
  float inv = rsqrtf(s / (float)D + 1e-6f);
  for (int i = threadIdx.x; i < D; i += blockDim.x) out[i] = w[i] * x[i] * inv;
}

// ---------------- GEMV: out[r] = (W[r,:].x + bias[r])*scale + res[r] ----------------
// one wave per row, float4 streaming loads (global_load_b128) + prefetch
__global__ void gemv_kernel(const float* __restrict__ W, const float* __restrict__ x,
                            const float* __restrict__ bias, const float* __restrict__ res,
                            float* __restrict__ out, int R, int K, float scale) {
  int wid  = (blockIdx.x * blockDim.x + threadIdx.x) >> 5;
  int lane = threadIdx.x & 31;
  if (wid >= R) return;
  const float4* Wr = (const float4*)(W + (size_t)wid * K);
  const float4* xv = (const float4*)x;
  int K4 = K >> 2;
  float acc = 0.f;
  for (int i = lane; i < K4; i += 32) {
    __builtin_prefetch(Wr + i + 128, 0, 0);   // ~2KB lookahead -> global_prefetch_b8
    float4 a = Wr[i], b = xv[i];
    acc += a.x * b.x + a.y * b.y + a.z * b.z + a.w * b.w;
  }
  acc = wave_reduce_sum(acc);
  if (lane == 0) {
    float r = acc;
    if (bias) r += bias[wid];
    r *= scale;
    if (res) r += res[wid];
    out[wid] = r;
  }
}

// ---------------- fused gate/up + SiLU ----------------
__global__ void mlp_gateup_kernel(const float* __restrict__ Wg, const float* __restrict__ Wu,
                                  const float* __restrict__ x, float* __restrict__ t) {
  int wid  = (blockIdx.x * blockDim.x + threadIdx.x) >> 5;
  int lane = threadIdx.x & 31;
  if (wid >= II) return;
  const float4* g4 = (const float4*)(Wg + (size_t)wid * D);
  const float4* u4 = (const float4*)(Wu + (size_t)wid * D);
  const float4* xv = (const float4*)x;
  float ag = 0.f, au = 0.f;
  for (int i = lane; i < D / 4; i += 32) {
    float4 xx = xv[i];
    float4 g = g4[i]; ag += g.x * xx.x + g.y * xx.y + g.z * xx.z + g.w * xx.w;
    float4 u = u4[i]; au += u.x * xx.x + u.y * xx.y + u.z * xx.z + u.w * xx.w;
  }
  ag = wave_reduce_sum(ag);
  au = wave_reduce_sum(au);
  if (lane == 0) t[wid] = (ag / (1.f + expf(-ag))) * au;   // silu(g)*u
}

// ---------------- RoPE on q/k, append k/v to caches ----------------
__global__ void rope_append_kernel(float* __restrict__ q, const float* __restrict__ kvec,
                                   const float* __restrict__ vvec,
                                   const float* __restrict__ cq, const float* __restrict__ sq,
                                   const float* __restrict__ ck, const float* __restrict__ sk,
                                   float* __restrict__ kcache, float* __restrict__ vcache) {
  int tid = threadIdx.x;
  // q: rotate_half on last dim (HD=128, 64 pairs per head)
  for (int i = tid; i < HQ * 64; i += blockDim.x) {
    int h = i >> 6, d = i & 63;
    float a = q[h * HDIM + d], b = q[h * HDIM + d + 64];
    q[h * HDIM + d]      = a * cq[d]      - b * sq[d];
    q[h * HDIM + d + 64] = b * cq[d + 64] + a * sq[d + 64];
  }
  // k: rotate_half on HD axis, write to last cache column (pos KVLEN-1)
  for (int i = tid; i < HKV * 64; i += blockDim.x) {
    int h = i >> 6, d = i & 63;
    float a = kvec[h * HDIM + d], b = kvec[h * HDIM + d + 64];
    float lo = a * ck[d]      - b * sk[d];
    float hi = b * ck[d + 64] + a * sk[d + 64];
    kcache[(size_t)(h * HDIM + d)      * KVLEN + (KVLEN - 1)] = lo;
    kcache[(size_t)(h * HDIM + d + 64) * KVLEN + (KVLEN - 1)] = hi;
  }
  // v: append row KVLEN-1
  for (int i = tid; i < HKV * HDIM; i += blockDim.x) {
    int h = i >> 7, d = i & 127;
    vcache[((size_t)h * KVLEN + (KVLEN - 1)) * HDIM + d] = vvec[i];
  }
}

// ---------------- attention scores via V_WMMA_F32_16X16X4_F32 ----------------
// scores[16, KVP] = Q[16,128] @ K[128, KVLEN]  (8 real heads per kv group, padded to 16)
// A layout (16x4 f32): lanes 0-15 row M, K=k0,k0+1 ; lanes 16-31 row M, K=k0+2,k0+3
// B layout (4x16):     lanes 0-15 col N with K rows (k0,k0+1); lanes 16-31 (k0+2,k0+3)
// Branch-free inner loop: padding rows read adjacent in-bounds ws garbage (their C
// rows are discarded); out-of-range columns are clamped (never stored).
__global__ void attn_scores_wmma(const float* __restrict__ qbuf,
                                 const float* __restrict__ kcache,
                                 float* __restrict__ scores) {
  int lane = threadIdx.x;
  int col0 = blockIdx.x * 16;
  int kv   = blockIdx.y;
  const float* K = kcache + (size_t)kv * HDIM * KVLEN;
  int m    = lane & 15;
  int koff = (lane >> 4) << 1;
  int n    = col0 + m;
  int nc   = (n < KVLEN) ? n : (KVLEN - 1);   // clamped column
  const float2* qr = (const float2*)(qbuf + (size_t)(kv * GQ + m) * HDIM + koff);
  v8f c = {};
#pragma unroll 4
  for (int k0 = 0; k0 < HDIM; k0 += 4) {
    float2 af = qr[k0 >> 1];                  // q[row][k0+koff .. +1], 8B aligned
    int kk = k0 + koff;
    v2f a, b;
    a.x = af.x; a.y = af.y;
    b.x = K[(size_t)kk * KVLEN + nc];
    b.y = K[(size_t)(kk + 1) * KVLEN + nc];
    c = wmma_f32_16x16x4(a, b, c);
  }
  // C layout: VGPR v, lanes 0-15 -> row M=v (our 8 real heads), col N=lane
  if (lane < 16 && n < KVLEN) {
#pragma unroll
    for (int v = 0; v < 8; ++v)
      scores[(size_t)(kv * GQ + v) * KVP + n] = c[v];
  }
}

// ---------------- softmax over KVLEN per head (exact two-pass) ----------------
__global__ void softmax_kernel(float* __restrict__ scores) {
  __shared__ float sm[8];
  float* row = scores + (size_t)blockIdx.x * KVP;
  float m = -3.4e38f;
  for (int i = threadIdx.x; i < KVLEN; i += blockDim.x) m = fmaxf(m, row[i]);
  m = block_reduce_max(m, sm);
  float s = 0.f;
  for (int i = threadIdx.x; i < KVLEN; i += blockDim.x) {
    float e = expf(row[i] - m); row[i] = e; s += e;
  }
  for (int i = KVLEN + threadIdx.x; i < KVP; i += blockDim.x) row[i] = 0.f;  // pad
  s = block_reduce_sum(s, sm);
  float inv = 1.f / s;
  for (int i = threadIdx.x; i < KVLEN; i += blockDim.x) row[i] *= inv;
}

// ---------------- attn @ V via V_WMMA_F32_16X16X4_F32, chunked over KV ----------------
// Branch-free: padding A rows clamp to row 7 (discarded in C); out-of-range KV rows
// clamp to KVLEN-1 while p's zero padding annihilates their contribution.
#define ACHUNK 132   // multiple of 4; 63 chunks cover 8196
__global__ void attn_out_wmma(const float* __restrict__ p, const float* __restrict__ vcache,
                              float* __restrict__ accum) {
  int lane  = threadIdx.x;
  int chunk = blockIdx.x, nt = blockIdx.y, kv = blockIdx.z;
  const float* V = vcache + (size_t)kv * KVLEN * HDIM;
  int m    = lane & 15;
  int koff = (lane >> 4) << 1;
  int n    = nt * 16 + m;
  int rowm = (m < 8) ? m : 7;                  // clamp padding rows
  const float2* pr = (const float2*)(p + (size_t)(kv * GQ + rowm) * KVP);
  v8f c = {};
  int kbeg = chunk * ACHUNK;
  int kend = kbeg + ACHUNK;
  if (kend > KLIM) kend = KLIM;
  for (int k0 = kbeg; k0 < kend; k0 += 4) {
    int kk = k0 + koff;                        // even -> 8B-aligned float2
    float2 af = pr[kk >> 1];                   // p pad [KVLEN,KVP) is zero
    int k1 = (kk     < KVLEN) ? kk     : (KVLEN - 1);
    int k2 = (kk + 1 < KVLEN) ? kk + 1 : (KVLEN - 1);
    v2f a, b;
    a.x = af.x; a.y = af.y;
    b.x = V[(size_t)k1 * HDIM + n];
    b.y = V[(size_t)k2 * HDIM + n];
    c = wmma_f32_16x16x4(a, b, c);
  }
  if (lane < 16) {
#pragma unroll
    for (int v = 0; v < 8; ++v)
      atomicAdd(&accum[(size_t)(kv * GQ + v) * HDIM + nt * 16 + lane], c[v]);
  }
}

// ---------------- LM head GEMV + fused argmax ----------------
__global__ void lm_argmax_kernel(const float* __restrict__ W, const float* __restrict__ x,
                                 unsigned long long* __restrict__ slot) {
  int wid  = (blockIdx.x * blockDim.x + threadIdx.x) >> 5;
  int lane = threadIdx.x & 31;
  if (wid >= VOCAB) return;
  const float4* Wr = (const float4*)(W + (size_t)wid * D);
  const float4* xv = (const float4*)x;
  float acc = 0.f;
  for (int i = lane; i < D / 4; i += 32) {
    __builtin_prefetch(Wr + i + 128, 0, 0);
    float4 a = Wr[i], b = xv[i];
    acc += a.x * b.x + a.y * b.y + a.z * b.z + a.w * b.w;
  }
  acc = wave_reduce_sum(acc);
  if (lane == 0) {
    unsigned u   = __float_as_uint(acc);
    unsigned ord = (u & 0x80000000u) ? ~u : (u | 0x80000000u);   // order-preserving
    unsigned long long key =
        ((unsigned long long)ord << 32) | (unsigned long long)(0xFFFFFFFFu - (unsigned)wid);
    atomicMax(slot, key);  // tie -> smallest index wins (matches jnp.argmax)
  }
}

__global__ void finalize_kernel(const unsigned long long* __restrict__ slot,
                                const int* __restrict__ kvlen, int* __restrict__ out_tail) {
  if (threadIdx.x == 0 && blockIdx.x == 0) {
    out_tail[0] = kvlen[0] + 1;
    unsigned long long k = *slot;
    out_tail[1] = (int)(0xFFFFFFFFu - (unsigned)(k & 0xFFFFFFFFull));
  }
}

// ---------------- host orchestration ----------------
extern "C" void kernel_launch(void* const* d_in, const int* in_sizes, int n_in,
                              void* d_out, int out_size, void* d_ws, size_t ws_size,
                              hipStream_t stream) {
  (void)in_sizes; (void)n_in; (void)out_size; (void)ws_size;

  const float* hidden = (const float*)d_in[0];
  const float* past_k = (const float*)d_in[1];
  const float* past_v = (const float*)d_in[2];
  const float* cq     = (const float*)d_in[3];
  const float* sq     = (const float*)d_in[4];
  const float* ck     = (const float*)d_in[5];
  const float* sk     = (const float*)d_in[6];
  const float* w_ln1  = (const float*)d_in[7];
  const float* wq     = (const float*)d_in[8];
  const float* bq     = (const float*)d_in[9];
  const float* wk     = (const float*)d_in[10];
  const float* bk     = (const float*)d_in[11];
  const float* wv     = (const float*)d_in[12];
  const float* bv     = (const float*)d_in[13];
  const float* wo     = (const float*)d_in[14];
  const float* w_ln2  = (const float*)d_in[15];
  const float* wg     = (const float*)d_in[16];
  const float* wu     = (const float*)d_in[17];
  const float* wd     = (const float*)d_in[18];
  const float* w_norm = (const float*)d_in[19];
  const float* w_lm   = (const float*)d_in[20];
  const int*   kvlen  = (const int*)d_in[21];

  // workspace layout (floats)
  float* ws   = (float*)d_ws;
  float* h    = ws;             // 2048
  float* xn   = ws + 2048;      // 2048
  float* q    = ws + 4096;      // 2048
  float* kvec = ws + 6144;      // 256
  float* vvec = ws + 6400;      // 256
  float* acc  = ws + 6656;      // 2048 (attn out, head-major = ao layout)
  float* t    = ws + 8704;      // 5504
  float* hl   = ws + 14208;     // 2048
  float* sc   = ws + 16384;     // 16 * KVP = 131328
  unsigned long long* slot = (unsigned long long*)(ws + 147712);  // 8B aligned

  // output layout: new_k | new_v | (int)kv_seq_len+1 | (int)max_ids
  float* outK = (float*)d_out;
  float* outV = outK + (size_t)NLAYER * HKV * HDIM * KVLEN;
  int* outTail = (int*)(outV + (size_t)NLAYER * HKV * KVLEN * HDIM);

  copy_kernel<<<(D + 255) / 256, 256, 0, stream>>>(hidden, h, D);
  copy_k_kernel<<<2048, 256, 0, stream>>>(past_k, outK);
  copy_v_kernel<<<2048, 256, 0, stream>>>((const float4*)past_v, (float4*)outV);

  for (int l = 0; l < NLAYER; ++l) {
    const float* wln1_l = w_ln1 + (size_t)l * D;
    const float* wq_l   = wq + (size_t)l * D * D;
    const float* bq_l   = bq + (size_t)l * D;
    const float* wk_l   = wk + (size_t)l * HKV * HDIM * D;
    const float* bk_l   = bk + (size_t)l * HKV * HDIM;
    const float* wv_l   = wv + (size_t)l * HKV * HDIM * D;
    const float* bv_l   = bv + (size_t)l * HKV * HDIM;
    const float* wo_l   = wo + (size_t)l * D * D;
    const float* wln2_l = w_ln2 + (size_t)l * D;
    const float* wg_l   = wg + (size_t)l * II * D;
    const float* wu_l   = wu + (size_t)l * II * D;
    const float* wd_l   = wd + (size_t)l * D * II;
    float* kc = outK + (size_t)l * HKV * HDIM * KVLEN;   // [HKV, HD, KVLEN]
    float* vc = outV + (size_t)l * HKV * KVLEN * HDIM;   // [HKV, KVLEN, HD]

    rmsnorm_kernel<<<1, 256, 0, stream>>>(h, wln1_l, xn);
    gemv_kernel<<<(D + 7) / 8, 256, 0, stream>>>(wq_l, xn, bq_l, nullptr, q, D, D, SCALEF);
    gemv_kernel<<<(HKV * HDIM + 7) / 8, 256, 0, stream>>>(wk_l, xn, bk_l, nullptr, kvec,
                                                          HKV * HDIM, D, SCALEF);
    gemv_kernel<<<(HKV * HDIM + 7) / 8, 256, 0, stream>>>(wv_l, xn, bv_l, nullptr, vvec,
                                                          HKV * HDIM, D, 1.0f);
    rope_append_kernel<<<1, 256, 0, stream>>>(q, kvec, vvec, cq, sq, ck, sk, kc, vc);

    zero_kernel<<<(HQ * HDIM + 255) / 256, 256, 0, stream>>>(acc, HQ * HDIM);
    attn_scores_wmma<<<dim3(KVP / 16, HKV), 32, 0, stream>>>(q, kc, sc);
    softmax_kernel<<<HQ, 256, 0, stream>>>(sc);
    attn_out_wmma<<<dim3((KLIM + ACHUNK - 1) / ACHUNK, HDIM / 16, HKV), 32, 0, stream>>>(sc, vc, acc);

    gemv_kernel<<<(D + 7) / 8, 256, 0, stream>>>(wo_l, acc, nullptr, h, h, D, D, 1.0f);
    rmsnorm_kernel<<<1, 256, 0, stream>>>(h, wln2_l, xn);
    mlp_gateup_kernel<<<(II + 7) / 8, 256, 0, stream>>>(wg_l, wu_l, xn, t);
    gemv_kernel<<<(D + 7) / 8, 256, 0, stream>>>(wd_l, t, nullptr, h, h, D, II, 1.0f);
  }

  rmsnorm_kernel<<<1, 256, 0, stream>>>(h, w_norm, hl);
  zero_kernel<<<1, 32, 0, stream>>>((float*)slot, 2);
  lm_argmax_kernel<<<(VOCAB + 7) / 8, 256, 0, stream>>>(w_lm, hl, slot);
  finalize_kernel<<<1, 32, 0, stream>>>(slot, kvlen, outTail);
}